// AUAvULoss_78769700209210
// MI455X (gfx1250) — compile-verified
//
#include <hip/hip_runtime.h>

typedef __attribute__((ext_vector_type(2))) float v2f;
typedef __attribute__((ext_vector_type(8))) float v8f;

#define N_ROWS 131072
#define N_COLS 1000
#define N_C4   250          // N_COLS / 4
#define N_TH   21
#define NB1    16384        // N_ROWS / 8  (per-block ce partials from k1)
#define NB2    256          // stage-2 reduction blocks
#define K4_BLOCKS 64
#define K4_WAVES (K4_BLOCKS * 8)   // 512 waves
#define SPW (N_ROWS / K4_WAVES)    // 256 samples per wave (divisible by 4)
#define BINS_PER_WAVE 128          // 32 threshold rows x 4 weight cols

// ---------------------------------------------------------------- k0: init
__global__ void k0_init(int* label_slot) { *label_slot = 0x7FFFFFFF; }

// -------------------------------------------- k1: per-row streaming pass
// one wave32 per row: max/argmax (first-occurrence), entropy, ce, label argmin
__global__ __launch_bounds__(256) void k1_rows(
    const float* __restrict__ probs, const float* __restrict__ y,
    float* __restrict__ conf, int* __restrict__ pred, float* __restrict__ unc,
    float* __restrict__ ce_part, int* __restrict__ label_slot) {
  const int wave = threadIdx.x >> 5;
  const int lane = threadIdx.x & 31;
  const int row  = blockIdx.x * 8 + wave;
  const float4* p4 = (const float4*)(probs + (size_t)row * N_COLS);
  const float4* y4 = (const float4*)(y     + (size_t)row * N_COLS);

  float vmax = -__builtin_inff();
  int   imax = 0x7FFFFFFF;
  float s_pl = 0.f, s_yl = 0.f;

  for (int c = lane; c < N_C4; c += 32) {
    const float4 p = p4[c];
    const float4 t = y4[c];
    const int cb = c << 2;
#define DO_ELEM(pv, yv, idx)                                        \
    { const float lg = logf(fmaxf((pv), 1e-10f));                   \
      s_pl += (pv) * lg; s_yl += (yv) * lg;                         \
      if ((pv) > vmax) { vmax = (pv); imax = (idx); }               \
      if ((yv) > 0.5f) atomicMin(label_slot, row * N_COLS + (idx)); }
    DO_ELEM(p.x, t.x, cb + 0)
    DO_ELEM(p.y, t.y, cb + 1)
    DO_ELEM(p.z, t.z, cb + 2)
    DO_ELEM(p.w, t.w, cb + 3)
#undef DO_ELEM
  }
  // wave32 reduction: max with first-index tiebreak, plus two sums
  for (int o = 16; o > 0; o >>= 1) {
    const float ov = __shfl_xor(vmax, o, 32);
    const int   oi = __shfl_xor(imax, o, 32);
    if (ov > vmax || (ov == vmax && oi < imax)) { vmax = ov; imax = oi; }
    s_pl += __shfl_xor(s_pl, o, 32);
    s_yl += __shfl_xor(s_yl, o, 32);
  }
  __shared__ float sh[8];
  if (lane == 0) {
    conf[row] = vmax; pred[row] = imax; unc[row] = -s_pl;
    sh[wave] = -s_yl;
  }
  __syncthreads();
  if (threadIdx.x == 0) {
    float s = 0.f;
    for (int i = 0; i < 8; i++) s += sh[i];
    ce_part[blockIdx.x] = s;
  }
}

// ------------------- k2: block partials of min/max(unc), T_au, T_iu, ce sum
__global__ __launch_bounds__(256) void k2_partial(
    const float* __restrict__ unc, const float* __restrict__ conf,
    const int* __restrict__ pred, const float* __restrict__ ce_part,
    const int* __restrict__ label_slot,
    float* __restrict__ minp, float* __restrict__ maxp,
    float* __restrict__ waup, float* __restrict__ wiup, float* __restrict__ cep) {
  const int tid = blockIdx.x * blockDim.x + threadIdx.x;
  const int stride = gridDim.x * blockDim.x;
  const int label = *label_slot;
  float lmin = __builtin_inff(), lmax = -__builtin_inff();
  float swau = 0.f, swiu = 0.f, sce = 0.f;
  for (int i = tid; i < N_ROWS; i += stride) {
    const float u = unc[i];
    lmin = fminf(lmin, u); lmax = fmaxf(lmax, u);
    const float tt = tanhf(u);
    const float cf = conf[i];
    if (pred[i] == label) swau += cf * tt;            // total w_au
    else                  swiu += (1.f - cf) * tt;    // total w_iu
  }
  for (int i = tid; i < NB1; i += stride) sce += ce_part[i];

  __shared__ float sA[256], sB[256], sC[256], sD[256], sE[256];
  const int t = threadIdx.x;
  sA[t] = lmin; sB[t] = lmax; sC[t] = swau; sD[t] = swiu; sE[t] = sce;
  __syncthreads();
  for (int o = 128; o > 0; o >>= 1) {
    if (t < o) {
      sA[t] = fminf(sA[t], sA[t + o]); sB[t] = fmaxf(sB[t], sB[t + o]);
      sC[t] += sC[t + o]; sD[t] += sD[t + o]; sE[t] += sE[t + o];
    }
    __syncthreads();
  }
  if (t == 0) {
    minp[blockIdx.x] = sA[0]; maxp[blockIdx.x] = sB[0];
    waup[blockIdx.x] = sC[0]; wiup[blockIdx.x] = sD[0]; cep[blockIdx.x] = sE[0];
  }
}

// ---------------------------------- k3: finalize scalars (single block)
__global__ __launch_bounds__(256) void k3_reduce(
    const float* __restrict__ minp, const float* __restrict__ maxp,
    const float* __restrict__ waup, const float* __restrict__ wiup,
    const float* __restrict__ cep, float* __restrict__ scal) {
  __shared__ float sA[256], sB[256], sC[256], sD[256], sE[256];
  const int t = threadIdx.x;
  sA[t] = minp[t]; sB[t] = maxp[t]; sC[t] = waup[t]; sD[t] = wiup[t]; sE[t] = cep[t];
  __syncthreads();
  for (int o = 128; o > 0; o >>= 1) {
    if (t < o) {
      sA[t] = fminf(sA[t], sA[t + o]); sB[t] = fmaxf(sB[t], sB[t + o]);
      sC[t] += sC[t + o]; sD[t] += sD[t + o]; sE[t] += sE[t + o];
    }
    __syncthreads();
  }
  if (t == 0) { scal[0]=sA[0]; scal[1]=sB[0]; scal[2]=sC[0]; scal[3]=sD[0]; scal[4]=sE[0]; }
}

// ------------- k4: threshold-bin GEMM  S[21,4] = le[21,N] @ W[N,4]  (WMMA)
// A (16x4 f32, 2 VGPRs): lane l -> row m=l&15, K = 2*(l>>4)+{0,1}
// B (4x16 f32, 2 VGPRs): lane l -> col n=l&15, K = 2*(l>>4)+{0,1}
// D (16x16 f32, 8 VGPRs): lane l, vgpr j -> row j+8*(l>>4), col l&15
__global__ __launch_bounds__(256) void k4_wmma(
    const float* __restrict__ unc, const float* __restrict__ conf,
    const int* __restrict__ pred, const float* __restrict__ scal,
    const int* __restrict__ label_slot, float* __restrict__ wave_bins) {
  const int lane = threadIdx.x & 31;
  const int gw   = (blockIdx.x * blockDim.x + threadIdx.x) >> 5;
  const float umin = scal[0];
  const float urng = scal[1] - scal[0];
  const int label = *label_slot;
  const int mn = lane & 15;            // threshold row (A) == weight col (B)
  const int kb = (lane >> 4) << 1;     // this lane's K pair {kb, kb+1}
  const float th0 = umin + ((float)mn * 0.05f) * urng;
  const float th1 = (mn + 16 < N_TH)
                      ? (umin + ((float)(mn + 16) * 0.05f) * urng)
                      : -__builtin_inff();   // padded rows -> indicator 0

  v8f c0 = {}; v8f c1 = {};
  const int sbeg = gw * SPW;
  for (int s = sbeg; s < sbeg + SPW; s += 4) {
    const int i0 = s + kb, i1 = i0 + 1;
    const float u0 = unc[i0], u1 = unc[i1];
    v2f a0, a1, b;
    a0.x = (u0 <= th0) ? 1.f : 0.f;  a0.y = (u1 <= th0) ? 1.f : 0.f;
    a1.x = (u0 <= th1) ? 1.f : 0.f;  a1.y = (u1 <= th1) ? 1.f : 0.f;
    // weight columns: 0=w_ac 1=w_au 2=w_ic 3=w_iu (branch-free selects)
    const float t0 = tanhf(u0), t1 = tanhf(u1);
    const float cf0 = conf[i0], cf1 = conf[i1];
    const bool acc0 = (pred[i0] == label), acc1 = (pred[i1] == label);
    const float tw0 = (mn & 1) ? t0 : (1.f - t0);
    const float tw1 = (mn & 1) ? t1 : (1.f - t1);
    const float wa0 = acc0 ? cf0 * tw0 : 0.f;          // cols 0/1
    const float wa1 = acc1 ? cf1 * tw1 : 0.f;
    const float wi0 = acc0 ? 0.f : (1.f - cf0) * tw0;  // cols 2/3
    const float wi1 = acc1 ? 0.f : (1.f - cf1) * tw1;
    b.x = (mn >= 4) ? 0.f : ((mn < 2) ? wa0 : wi0);
    b.y = (mn >= 4) ? 0.f : ((mn < 2) ? wa1 : wi1);
    // D = A x B + C, fp32 exact (matches reference f32 matmul)
    c0 = __builtin_amdgcn_wmma_f32_16x16x4_f32(false, a0, false, b,
                                               (short)0, c0, false, false);
    c1 = __builtin_amdgcn_wmma_f32_16x16x4_f32(false, a1, false, b,
                                               (short)0, c1, false, false);
  }
  // spill this wave's 21x4 partial bins (rows 0..15 from tile0, 16..20 from tile1)
  float* myb = wave_bins + (size_t)gw * BINS_PER_WAVE;
  if (mn < 4) {
#pragma unroll
    for (int j = 0; j < 8; j++) {
      const int r0 = j + ((lane >> 4) << 3);
      myb[r0 * 4 + mn] = c0[j];
      const int r1 = 16 + r0;
      if (r1 < N_TH) myb[r1 * 4 + mn] = c1[j];
    }
  }
}

// ------------------------- k5: deterministic epilogue (single thread)
__global__ void k5_final(const float* __restrict__ wave_bins,
                         const float* __restrict__ scal,
                         float* __restrict__ out) {
  if (threadIdx.x != 0 || blockIdx.x != 0) return;
  float S[N_TH][4];
  for (int k = 0; k < N_TH; k++)
    for (int j = 0; j < 4; j++) S[k][j] = 0.f;
  for (int w = 0; w < K4_WAVES; w++) {
    const float* b = wave_bins + (size_t)w * BINS_PER_WAVE;
    for (int k = 0; k < N_TH; k++)
      for (int j = 0; j < 4; j++) S[k][j] += b[k * 4 + j];
  }
  const float Tau = scal[2], Tiu = scal[3];
  const float ce  = scal[4] / (float)N_ROWS;
  float avu[N_TH];
  for (int k = 0; k < N_TH; k++) {
    const float n_ac = S[k][0];
    const float n_au = Tau - S[k][1];   // gt @ w_au = total - le @ w_au
    const float n_ic = S[k][2];
    const float n_iu = Tiu - S[k][3];
    avu[k] = (n_ac + n_iu) / (n_ac + n_au + n_ic + n_iu + 1e-10f);
  }
  float auc = 0.f;
  for (int k = 0; k < N_TH - 1; k++) {
    const float dx = (float)(k + 1) * 0.05f - (float)k * 0.05f;
    auc += (avu[k + 1] + avu[k]) * 0.5f * dx;
  }
  out[0] = -logf(auc + 1e-10f) + ce;   // BETA = 1
  out[1] = ce;
}

// ---------------------------------------------------------------- launch
extern "C" void kernel_launch(void* const* d_in, const int* in_sizes, int n_in,
                              void* d_out, int out_size, void* d_ws, size_t ws_size,
                              hipStream_t stream) {
  const float* probs = (const float*)d_in[0];
  const float* y     = (const float*)d_in[1];
  float* out = (float*)d_out;

  // workspace carve-out (~1.9 MB)
  float* conf      = (float*)d_ws;                  // N_ROWS
  int*   pred      = (int*)(conf + N_ROWS);         // N_ROWS
  float* unc       = (float*)(pred + N_ROWS);       // N_ROWS
  float* ce_part   = unc + N_ROWS;                  // NB1
  float* minp      = ce_part + NB1;                 // NB2
  float* maxp      = minp + NB2;                    // NB2
  float* waup      = maxp + NB2;                    // NB2
  float* wiup      = waup + NB2;                    // NB2
  float* cep       = wiup + NB2;                    // NB2
  float* scal      = cep + NB2;                     // 5 scalars
  int*   label     = (int*)(scal + 5);              // 1 (+2 pad)
  float* wave_bins = (float*)(label + 3);           // K4_WAVES * 128

  k0_init<<<1, 1, 0, stream>>>(label);
  k1_rows<<<N_ROWS / 8, 256, 0, stream>>>(probs, y, conf, pred, unc, ce_part, label);
  k2_partial<<<NB2, 256, 0, stream>>>(unc, conf, pred, ce_part, label,
                                      minp, maxp, waup, wiup, cep);
  k3_reduce<<<1, 256, 0, stream>>>(minp, maxp, waup, wiup, cep, scal);
  k4_wmma<<<K4_BLOCKS, 256, 0, stream>>>(unc, conf, pred, scal, label, wave_bins);
  k5_final<<<1, 1, 0, stream>>>(wave_bins, scal, out);
}